// Neg_64510408785972
// MI455X (gfx1250) — compile-verified
//
#include <hip/hip_runtime.h>
#include <hip/hip_bf16.h>

// ---------------------------------------------------------------------------
// Problem: N=1024, TOPK=16, B=64, W=32.
//   topk idx[1024,16]  ->  gather-GEMM:  y[b,o,w] = sum_{c,kh} X * conv_w + bias
//   GEMM dims: M=(b,w)=2048, N=o=1024, K=(c,kh)=16384, fp32.
// ---------------------------------------------------------------------------

typedef __attribute__((ext_vector_type(2)))  float v2f;
typedef __attribute__((ext_vector_type(8)))  float v8f;
typedef __attribute__((ext_vector_type(4)))  unsigned int u32x4;
typedef __attribute__((ext_vector_type(8)))  int          i32x8;
typedef __attribute__((ext_vector_type(4)))  int          i32x4;

#define NDIM   1024
#define TOPK   16
#define KSUM   16384      // 1024 channels * 16 kh
#define O_TILE 128        // output-channel tile per workgroup
#define K_TILE 64         // K-slice per iteration
#define ROWP   66         // LDS row pitch in floats (64 + 2 pad dwords)

// ---------------------------------------------------------------------------
// Kernel 1: per-row top-16 (descending, lowest-index tiebreak). One wave/row.
// ---------------------------------------------------------------------------
__global__ __launch_bounds__(32) void topk_kernel(const float* __restrict__ f1,
                                                  int* __restrict__ topk_idx) {
  const int row  = blockIdx.x;
  const int lane = threadIdx.x;
  const float* rp = f1 + (size_t)row * NDIM;

  float v[32];
#pragma unroll
  for (int j = 0; j < 32; ++j) v[j] = rp[lane + 32 * j];

  for (int t = 0; t < TOPK; ++t) {
    float bv = v[0];
    int   bi = lane;
#pragma unroll
    for (int j = 1; j < 32; ++j) {
      float x = v[j];
      if (x > bv) { bv = x; bi = lane + 32 * j; }   // strict > keeps lowest idx
    }
    // wave32 butterfly reduce (max value, ties -> lower index)
#pragma unroll
    for (int off = 16; off > 0; off >>= 1) {
      float ov = __shfl_xor(bv, off, 32);
      int   oi = __shfl_xor(bi, off, 32);
      if (ov > bv || (ov == bv && oi < bi)) { bv = ov; bi = oi; }
    }
    if (lane == 0) topk_idx[row * TOPK + t] = bi;
    // mask out the winner (branchless, keeps v[] in registers)
#pragma unroll
    for (int j = 0; j < 32; ++j)
      if (lane + 32 * j == bi) v[j] = -__builtin_inff();
  }
}

// ---------------------------------------------------------------------------
// Kernel 2: gather-GEMM with fp32 WMMA.
//   grid = (8 o-tiles, 64 b).  block = 256 (8 waves).
//   wave w owns output [m=0..31] x [o = obase + 16w .. +15]  -> two 16x16 C tiles.
//   W tile [O_TILE x K_TILE] streamed by the Tensor Data Mover into LDS
//   (pad_enable -> 66-dword pitch). X tile gathered by all threads.
// ---------------------------------------------------------------------------
__global__ __launch_bounds__(256) void gemm_kernel(const float* __restrict__ f1,
                                                   const float* __restrict__ conv_w,
                                                   const float* __restrict__ conv_b,
                                                   const int*   __restrict__ topk_idx,
                                                   float* __restrict__ out) {
  __shared__ float w_lds[O_TILE * ROWP];   // [o_local][kk]  pitch 66
  __shared__ float x_lds[32 * ROWP];       // [w_pix][kk]    pitch 66

  const int tid   = threadIdx.x;
  const int lane  = tid & 31;
  const int wave  = tid >> 5;
  const int obase = blockIdx.x * O_TILE;
  const int b     = blockIdx.y;

  v8f c0 = {};   // m = 0..15
  v8f c1 = {};   // m = 16..31

  const unsigned long long wbase = (unsigned long long)(uintptr_t)conv_w;
  const unsigned int lds_w_off   = (unsigned int)(uintptr_t)&w_lds[0];

  for (int k0 = 0; k0 < KSUM; k0 += K_TILE) {
    __syncthreads();   // previous tile fully consumed

    // ---- TDM: stream W'[obase..obase+127][k0..k0+63] into LDS (wave 0 issues)
    if (wave == 0) {
      unsigned long long ga = wbase + 4ull * ((unsigned long long)obase * KSUM + (unsigned)k0);
      u32x4 g0;
      g0.x = 1u;                                   // count=1, user descriptor
      g0.y = lds_w_off;                            // LDS dest
      g0.z = (unsigned int)ga;                     // global_addr[31:0]
      g0.w = (unsigned int)((ga >> 32) & 0x1FFFFFFull) | 0x80000000u; // addr[56:32] | type=2
      i32x8 g1;
      g1[0] = (int)((2u << 16)          // data_size = 4B
                  | (1u << 20)          // pad_enable
                  | (5u << 22)          // pad_interval: every 64 dwords
                  | (1u << 25));        // pad_amount: 2 dwords  -> pitch 66
      g1[1] = (int)(((unsigned)KSUM & 0xFFFFu) << 16);   // tensor_dim0 lo (16384)
      g1[2] = (int)(((unsigned)KSUM >> 16) | (1024u << 16)); // dim0 hi | tensor_dim1 lo
      g1[3] = (int)((unsigned)K_TILE << 16);             // dim1 hi=0 | tile_dim0 = 64
      g1[4] = (int)(unsigned)O_TILE;                     // tile_dim1 = 128, tile_dim2 = 0
      g1[5] = KSUM;                                      // tensor_dim0_stride lo
      g1[6] = 0;
      g1[7] = 0;
      i32x4 gz4 = {0, 0, 0, 0};                          // groups 2/3 unused (2D tensor)
      i32x8 gz8 = {0, 0, 0, 0, 0, 0, 0, 0};
      __builtin_amdgcn_tensor_load_to_lds(g0, g1, gz4, gz4, gz8, 0);
    }

    // ---- gather X tile: kappa = k0+kk -> (c,kh); rep half or topk-gather half
#pragma unroll
    for (int pass = 0; pass < 2; ++pass) {
      const int kk  = (tid >> 3) + pass * 32;        // 0..63
      const int w4  = (tid & 7) * 4;                 // 0,4,...,28
      const int kap = k0 + kk;
      const int c   = kap >> 4;
      const int kh  = kap & 15;
      const int np  = c >> 6;
      const int k   = (c >> 2) & 15;
      const int mhi = c & 3;
      const int n   = 16 * b + np;
      const int row = (mhi < 2) ? n : topk_idx[n * TOPK + k];
      const int col = (mhi & 1) * 512 + kh * 32 + w4;
      const float4 val = *(const float4*)(f1 + (size_t)row * NDIM + col);
      x_lds[(w4 + 0) * ROWP + kk] = val.x;
      x_lds[(w4 + 1) * ROWP + kk] = val.y;
      x_lds[(w4 + 2) * ROWP + kk] = val.z;
      x_lds[(w4 + 3) * ROWP + kk] = val.w;
    }

    if (wave == 0) __builtin_amdgcn_s_wait_tensorcnt(0);
    __syncthreads();

    // ---- 16 steps of V_WMMA_F32_16X16X4_F32 per C tile
    // A-frag lane layout (16x4 f32): M = lane%16, K = kk + 2*(lane/16) + v
    // B-frag:                        N = lane%16, K = kk + 2*(lane/16) + v
    const int koff = 2 * (lane >> 4);
    const float* bp  = &w_lds[(wave * 16 + (lane & 15)) * ROWP + koff];
    const float* a0p = &x_lds[(lane & 15) * ROWP + koff];
    const float* a1p = &x_lds[(16 + (lane & 15)) * ROWP + koff];
#pragma unroll
    for (int kk = 0; kk < K_TILE; kk += 4) {
      const v2f bf = *(const v2f*)(bp  + kk);
      const v2f a0 = *(const v2f*)(a0p + kk);
      const v2f a1 = *(const v2f*)(a1p + kk);
      c0 = __builtin_amdgcn_wmma_f32_16x16x4_f32(false, a0, false, bf, (short)0, c0, false, false);
      c1 = __builtin_amdgcn_wmma_f32_16x16x4_f32(false, a1, false, bf, (short)0, c1, false, false);
    }
  }

  // ---- epilogue: bias + relu, scatter per output reshape [1024, 2048]
  // C/D layout: N = lane%16, M = 8*(lane/16) + j
  const int og   = obase + wave * 16 + (lane & 15);
  const float bias = conv_b[og];
  const int r    = 16 * b + (og >> 6);
  const int cb   = (og & 63) * 32;
  float* orow    = out + (size_t)r * 2048 + cb;
  const int mb   = 8 * (lane >> 4);
#pragma unroll
  for (int j = 0; j < 8; ++j) {
    float v0 = c0[j] + bias;
    float v1 = c1[j] + bias;
    orow[mb + j]      = v0 > 0.f ? v0 : 0.f;
    orow[16 + mb + j] = v1 > 0.f ? v1 : 0.f;
  }
}

// ---------------------------------------------------------------------------
extern "C" void kernel_launch(void* const* d_in, const int* in_sizes, int n_in,
                              void* d_out, int out_size, void* d_ws, size_t ws_size,
                              hipStream_t stream) {
  const float* f1     = (const float*)d_in[0];   // [1024,1024]
  const float* conv_w = (const float*)d_in[1];   // [1024,1024,16,1]
  const float* conv_b = (const float*)d_in[2];   // [1024]
  float* out = (float*)d_out;                    // [1024,2048]
  int* topk_idx = (int*)d_ws;                    // 1024*16 ints = 64 KB

  topk_kernel<<<NDIM, 32, 0, stream>>>(f1, topk_idx);
  gemm_kernel<<<dim3(NDIM / O_TILE, 64), 256, 0, stream>>>(f1, conv_w, conv_b, topk_idx, out);
}